// HMQAct_59983513256180
// MI455X (gfx1250) — compile-verified
//
#include <hip/hip_runtime.h>
#include <stdint.h>

// ---------------------------------------------------------------------------
// BFP quantization (block=16, shared exponent) for MI455X / gfx1250.
// Streaming, HBM-bound kernel (~512 MiB @ 23.3 TB/s ~= 23 us floor).
// CDNA5 paths: async global->LDS copies (ASYNCcnt) for double-buffered
// prefetch, ds_load_b128 readback, DPP quad_perm cross-lane max,
// non-temporal b128 stores.
// ---------------------------------------------------------------------------

typedef float v4f __attribute__((ext_vector_type(4)));
typedef __attribute__((address_space(3))) char lds_char_t;

#define TPB 256                 // 8 wave32s
#define FLOATS_PER_TILE 1024    // TPB * 4 floats; 4 KB per tile
#define V4_PER_TILE 256

// Raw LDS byte offset (addrspace(3) ptrtoint) for the async-copy VDST operand.
static __device__ __forceinline__ uint32_t lds_byte_off(const void* p) {
  return (uint32_t)(uintptr_t)(lds_char_t*)p;
}

// CDNA5 async copy: 16 bytes per lane, global -> LDS, tracked by ASYNCcnt.
static __device__ __forceinline__ void async_copy_b128(uint32_t lds_off,
                                                       const void* gptr) {
  asm volatile("global_load_async_to_lds_b128 %0, %1, off"
               :
               : "v"(lds_off), "v"(gptr)
               : "memory");
}

#define S_WAIT_ASYNCCNT(N) \
  asm volatile("s_wait_asynccnt %0" ::"i"(N) : "memory")

// DPP quad_perm lane exchange (pure VALU; replaces ds_bpermute shuffles).
// CTRL: 0xB1 = [1,0,3,2] (xor 1), 0x4E = [2,3,0,1] (xor 2).
template <int CTRL>
static __device__ __forceinline__ float dpp_qperm(float v) {
  int i = __builtin_bit_cast(int, v);
  int r = __builtin_amdgcn_update_dpp(i, i, CTRL, 0xF, 0xF, true);
  return __builtin_bit_cast(float, r);
}

// Quantize one value given up = (mb-1) - floor(log2(blockmax)).
static __device__ __forceinline__ float quant1(float x, int up, float qmin,
                                               float qmax) {
  float q = __builtin_rintf(__builtin_ldexpf(x, up));   // x / step, RNE
  q = fminf(fmaxf(q, qmin), qmax);                      // clip AFTER round
  return __builtin_ldexpf(q, -up);                      // * step
}

__global__ __launch_bounds__(TPB) void bfp_quant_kernel(
    const float* __restrict__ x, const int* __restrict__ mb_ptr,
    float* __restrict__ out, long long n) {
  __shared__ v4f stage[2][TPB];

  const int tid = threadIdx.x;
  const int mb = *mb_ptr;                      // mantissa bits (8)
  const int sh = mb - 1;                       // 7
  const float qmax = (float)((1 << sh) - 1);   // 127
  const float qmin = -(float)(1 << sh);        // -128

  const long long ntiles = n / FLOATS_PER_TILE;
  long long tile = blockIdx.x;
  const long long tstride = gridDim.x;

  const uint32_t lds0 = lds_byte_off(&stage[0][tid]);
  const uint32_t lds1 = lds_byte_off(&stage[1][tid]);

  if (tile < ntiles) {
    // Prologue: fetch first tile into buffer 0.
    async_copy_b128(lds0, x + tile * FLOATS_PER_TILE + (long long)tid * 4);
    int buf = 0;
    for (;;) {
      const long long next = tile + tstride;
      const bool have_next = next < ntiles;   // uniform across the block
      if (have_next) {
        // Prefetch tile t+1 into the other buffer while we process tile t.
        async_copy_b128(buf ? lds0 : lds1,
                        x + next * FLOATS_PER_TILE + (long long)tid * 4);
        S_WAIT_ASYNCCNT(1);   // in-order completion => current tile resident
      } else {
        S_WAIT_ASYNCCNT(0);
      }

      v4f v = stage[buf][tid];                 // ds_load_b128

      // Block max over 16 elements = 4 values/lane x 4 aligned lanes.
      float m = fmaxf(fmaxf(fabsf(v.x), fabsf(v.y)),
                      fmaxf(fabsf(v.z), fabsf(v.w)));
      m = fmaxf(m, dpp_qperm<0xB1>(m));        // xor 1 within quad
      m = fmaxf(m, dpp_qperm<0x4E>(m));        // xor 2 within quad

      // e = floor(log2(max(m, 1e-38))): frexp gives mant in [0.5,1) => -1.
      const int e = __builtin_amdgcn_frexp_expf(fmaxf(m, 1e-38f)) - 1;
      const int up = sh - e;

      v4f q;
      q.x = quant1(v.x, up, qmin, qmax);
      q.y = quant1(v.y, up, qmin, qmax);
      q.z = quant1(v.z, up, qmin, qmax);
      q.w = quant1(v.w, up, qmin, qmax);

      // Stream-once output: non-temporal b128 store.
      __builtin_nontemporal_store(q,
          (v4f*)out + tile * V4_PER_TILE + tid);

      if (!have_next) break;
      tile = next;
      buf ^= 1;
    }
  }

  // ---- Scalar tail (n % 1024 != 0; dead for the 4x4096x4096 shape) ----
  const long long total_blk = n / 16;
  const long long done_blk = ntiles * (FLOATS_PER_TILE / 16);
  const long long bidx = done_blk + (long long)blockIdx.x * TPB + tid;
  if (bidx < total_blk) {
    const float* p = x + bidx * 16;
    float vals[16];
    float m = 0.0f;
    #pragma unroll
    for (int i = 0; i < 16; ++i) {
      vals[i] = p[i];
      m = fmaxf(m, fabsf(vals[i]));
    }
    const int e = __builtin_amdgcn_frexp_expf(fmaxf(m, 1e-38f)) - 1;
    const int up = sh - e;
    float* o = out + bidx * 16;
    #pragma unroll
    for (int i = 0; i < 16; ++i) o[i] = quant1(vals[i], up, qmin, qmax);
  }
}

extern "C" void kernel_launch(void* const* d_in, const int* in_sizes, int n_in,
                              void* d_out, int out_size, void* d_ws,
                              size_t ws_size, hipStream_t stream) {
  const float* x = (const float*)d_in[0];
  const int* mb = (const int*)d_in[1];       // mantissa_bits (scalar)
  // d_in[2] = block_size; layout (4 lanes/block, 16 elems) assumes 16.
  float* out = (float*)d_out;

  const long long n = (long long)in_sizes[0];
  const long long ntiles = n / FLOATS_PER_TILE;

  long long grid = ntiles;
  if (grid > 4096) grid = 4096;   // ~16 tiles/block: amortizes pipeline fill
  if (grid < 1) grid = 1;         // tail-only case

  bfp_quant_kernel<<<(int)grid, TPB, 0, stream>>>(x, mb, out, n);
}